// AdvancedGNNModel_45011257262391
// MI455X (gfx1250) — compile-verified
//
#include <hip/hip_runtime.h>
#include <hip/hip_bf16.h>
#include <math.h>

// ---------------------------------------------------------------------------
// GAT x3 + mean-pool + MLP for MI455X (gfx1250, wave32).
// Dense math (feature transforms) goes through V_WMMA_F32_16X16X4_F32.
// Edge softmax/aggregation is HBM-bound scatter/gather with f32 atomics.
// ---------------------------------------------------------------------------

typedef float v2f __attribute__((ext_vector_type(2)));
typedef float v8f __attribute__((ext_vector_type(8)));

#define NHEADS 4
#define NEG_SLOPE 0.2f
#define MT 4   // M-tiles (16 rows each) per wave in the WMMA GEMM

// Monotonic float<->uint mapping so segment-max can use atomicMax(u32).
__device__ __forceinline__ unsigned f2mono(float f) {
  unsigned u = __float_as_uint(f);
  return (u & 0x80000000u) ? ~u : (u | 0x80000000u);
}
__device__ __forceinline__ float mono2f(unsigned u) {
  return (u & 0x80000000u) ? __uint_as_float(u & 0x7fffffffu)
                           : __uint_as_float(~u);
}

// ---------------------------------------------------------------------------
// C[M,Nc] = A[M,K] @ B[K,Nc], fp32 via WMMA 16x16x4.
// Block = 4 waves; each wave owns one 16-column tile and MT=4 row tiles
// (64 rows), so each B fragment (2 strided b32 loads) feeds 4 WMMAs.
// Out-of-range A rows are clamped to row 0: in WMMA an A row only feeds the
// matching D row, which is masked at the store, so no EXEC juggling needed.
// ---------------------------------------------------------------------------
__global__ __launch_bounds__(128) void gemm_f32_wmma(
    const float* __restrict__ A, const float* __restrict__ B,
    float* __restrict__ C, int M, int K, int Nc) {
  const int lane  = threadIdx.x & 31;
  const int wave  = threadIdx.x >> 5;
  const int nTile = blockIdx.y * 4 + wave;
  if (nTile * 16 >= Nc) return;                   // wave-uniform guard

  const int m0   = blockIdx.x * (16 * MT);
  const int colQ = nTile * 16 + (lane & 15);      // requested column
  const bool cOk = colQ < Nc;
  const int col  = cOk ? colQ : (Nc - 1);         // clamped (store masked)
  const int kh   = (lane >> 4) << 1;              // 0 or 2

  const float* ap[MT];
#pragma unroll
  for (int t = 0; t < MT; ++t) {
    int row = m0 + t * 16 + (lane & 15);
    ap[t] = A + (size_t)(row < M ? row : 0) * K + kh;  // clamp, no divergence
  }
  const float* bp = B + (size_t)kh * Nc + col;
  const size_t bstep = (size_t)4 * Nc;            // induction: one add/iter

  v8f acc[MT];
#pragma unroll
  for (int t = 0; t < MT; ++t) acc[t] = (v8f){};

#pragma unroll 2
  for (int k = 0; k < K; k += 4) {
    v2f b;
    b.x = bp[0];
    b.y = bp[Nc];
    bp += bstep;
#pragma unroll
    for (int t = 0; t < MT; ++t) {
      v2f a = *(const v2f*)(ap[t] + k);           // contiguous 8B load
      acc[t] = __builtin_amdgcn_wmma_f32_16x16x4_f32(
          false, a, false, b, (short)0, acc[t], false, false);
    }
  }

  // D layout: VGPR r -> M = r + 8*(lane>=16), N = lane%16
  const int half = (lane >> 4) << 3;
#pragma unroll
  for (int t = 0; t < MT; ++t) {
    int mBase = m0 + t * 16 + half;
#pragma unroll
    for (int r = 0; r < 8; ++r) {
      int m = mBase + r;
      if (m < M && cOk) C[(size_t)m * Nc + colQ] = acc[t][r];
    }
  }
}

// ---------------------------------------------------------------------------
// Per (node, head): s = <hW, a_src>, d = <hW, a_dst> over HID channels.
// One wave per (node, head).
// ---------------------------------------------------------------------------
__global__ void attn_sd(const float* __restrict__ hW,
                        const float* __restrict__ a_src,
                        const float* __restrict__ a_dst,
                        float* __restrict__ s, float* __restrict__ d,
                        int N, int HID) {
  int gw   = (int)((blockIdx.x * (size_t)blockDim.x + threadIdx.x) >> 5);
  int lane = threadIdx.x & 31;
  int node = gw / NHEADS;
  int head = gw - node * NHEADS;
  if (node >= N) return;
  const float* hp = hW + (size_t)node * NHEADS * HID + (size_t)head * HID;
  const float* as = a_src + (size_t)head * HID;
  const float* ad = a_dst + (size_t)head * HID;
  float ss = 0.f, dd = 0.f;
  for (int c = lane; c < HID; c += 32) {
    float v = hp[c];
    ss += v * as[c];
    dd += v * ad[c];
  }
#pragma unroll
  for (int off = 16; off; off >>= 1) {
    ss += __shfl_down(ss, off, 32);
    dd += __shfl_down(dd, off, 32);
  }
  if (lane == 0) {
    s[node * NHEADS + head] = ss;
    d[node * NHEADS + head] = dd;
  }
}

// agg = 0 ; mbits = mono(-inf) ; denom = 0
__global__ void init_layer(float* __restrict__ agg, unsigned* __restrict__ mbits,
                           float* __restrict__ denom, size_t nC, size_t nH) {
  size_t i = blockIdx.x * (size_t)blockDim.x + threadIdx.x;
  if (i < nC) agg[i] = 0.f;
  if (i < nH) { mbits[i] = f2mono(-INFINITY); denom[i] = 0.f; }
}

// Pass 1: e = leaky_relu(s[src]+d[dst]); segment max via monotonic atomicMax.
// Edge ids >= E are implicit self-loops (src=dst=eid-E).
__global__ void edge_p1(const int* __restrict__ src, const int* __restrict__ dst,
                        const float* __restrict__ s, const float* __restrict__ d,
                        float* __restrict__ e, unsigned* __restrict__ mbits,
                        int E, int N) {
  int t = (int)(blockIdx.x * (size_t)blockDim.x + threadIdx.x);
  int eid = t >> 2, h = t & 3;
  if (eid >= E + N) return;
  int u = eid < E ? src[eid] : eid - E;
  int v = eid < E ? dst[eid] : eid - E;
  float x = s[u * NHEADS + h] + d[v * NHEADS + h];
  x = x > 0.f ? x : NEG_SLOPE * x;
  e[(size_t)eid * NHEADS + h] = x;
  atomicMax(&mbits[v * NHEADS + h], f2mono(x));
}

// Pass 2: e <- exp(e - max[dst]); denom[dst] += e
__global__ void edge_p2(const int* __restrict__ dst, float* __restrict__ e,
                        const unsigned* __restrict__ mbits,
                        float* __restrict__ denom, int E, int N) {
  int t = (int)(blockIdx.x * (size_t)blockDim.x + threadIdx.x);
  int eid = t >> 2, h = t & 3;
  if (eid >= E + N) return;
  int v = eid < E ? dst[eid] : eid - E;
  float m  = mono2f(mbits[v * NHEADS + h]);
  float ex = __expf(e[(size_t)eid * NHEADS + h] - m);
  e[(size_t)eid * NHEADS + h] = ex;
  atomicAdd(&denom[v * NHEADS + h], ex);
}

// Pass 3: agg[dst] += hW[src] * alpha   (one block per edge, C channels)
__global__ void edge_p3(const int* __restrict__ src, const int* __restrict__ dst,
                        const float* __restrict__ hW, const float* __restrict__ e,
                        const float* __restrict__ denom, float* __restrict__ agg,
                        int E, int N, int C, int HID) {
  int eid = blockIdx.x;
  int u = eid < E ? src[eid] : eid - E;
  int v = eid < E ? dst[eid] : eid - E;
  const float* hrow = hW + (size_t)u * C;
  float* arow = agg + (size_t)v * C;
  for (int c = threadIdx.x; c < C; c += blockDim.x) {
    int h = c / HID;
    float alpha = e[(size_t)eid * NHEADS + h] /
                  (denom[v * NHEADS + h] + 1e-16f);
    atomicAdd(&arow[c], hrow[c] * alpha);
  }
}

// agg <- relu(agg + b) (+ residual)   (in place)
__global__ void finish_layer(float* __restrict__ agg, const float* __restrict__ b,
                             const float* __restrict__ res, size_t nC, int C) {
  size_t i = blockIdx.x * (size_t)blockDim.x + threadIdx.x;
  if (i >= nC) return;
  float v = agg[i] + b[i % C];
  v = v > 0.f ? v : 0.f;
  agg[i] = res ? v + res[i] : v;
}

__global__ void zero_f(float* __restrict__ p, size_t n) {
  size_t i = blockIdx.x * (size_t)blockDim.x + threadIdx.x;
  if (i < n) p[i] = 0.f;
}

__global__ void pool_cnt(const int* __restrict__ batch, float* __restrict__ cnt,
                         int N) {
  int n = (int)(blockIdx.x * (size_t)blockDim.x + threadIdx.x);
  if (n < N) atomicAdd(&cnt[batch[n]], 1.f);
}

__global__ void pool_add(const float* __restrict__ h, const int* __restrict__ batch,
                         float* __restrict__ sums, int N, int C) {
  size_t i = blockIdx.x * (size_t)blockDim.x + threadIdx.x;
  if (i >= (size_t)N * C) return;
  int n = (int)(i / C), c = (int)(i - (size_t)n * C);
  atomicAdd(&sums[(size_t)batch[n] * C + c], h[i]);
}

__global__ void pool_div(const float* __restrict__ sums, const float* __restrict__ cnt,
                         float* __restrict__ hp, int G, int C) {
  int i = (int)(blockIdx.x * (size_t)blockDim.x + threadIdx.x);
  if (i >= G * C) return;
  int g = i / C;
  float c = cnt[g];
  hp[i] = sums[i] / (c > 1.f ? c : 1.f);
}

// out[g] = relu(tmp[g,:] + cb1) . cW2 + cb2   (one wave per graph)
__global__ void mlp_tail(const float* __restrict__ tmp, const float* __restrict__ cb1,
                         const float* __restrict__ cW2, const float* __restrict__ cb2,
                         float* __restrict__ out, int HID) {
  int g = blockIdx.x, lane = threadIdx.x;
  float acc = 0.f;
  for (int c = lane; c < HID; c += 32) {
    float v = tmp[(size_t)g * HID + c] + cb1[c];
    v = v > 0.f ? v : 0.f;
    acc += v * cW2[c];
  }
#pragma unroll
  for (int off = 16; off; off >>= 1) acc += __shfl_down(acc, off, 32);
  if (lane == 0) out[g] = acc + cb2[0];
}

// ---------------------------------------------------------------------------
extern "C" void kernel_launch(void* const* d_in, const int* in_sizes, int n_in,
                              void* d_out, int out_size, void* d_ws, size_t ws_size,
                              hipStream_t stream) {
  (void)n_in; (void)out_size; (void)ws_size;
  const float* x     = (const float*)d_in[0];
  const int*   ei    = (const int*)d_in[1];
  const int*   batch = (const int*)d_in[2];
  const float* W[3]    = {(const float*)d_in[3],  (const float*)d_in[7],  (const float*)d_in[11]};
  const float* aS[3]   = {(const float*)d_in[4],  (const float*)d_in[8],  (const float*)d_in[12]};
  const float* aD[3]   = {(const float*)d_in[5],  (const float*)d_in[9],  (const float*)d_in[13]};
  const float* bias[3] = {(const float*)d_in[6],  (const float*)d_in[10], (const float*)d_in[14]};
  const float* cW1 = (const float*)d_in[15];
  const float* cb1 = (const float*)d_in[16];
  const float* cW2 = (const float*)d_in[17];
  const float* cb2 = (const float*)d_in[18];

  const int N    = in_sizes[2];
  const int Fin  = in_sizes[0] / N;
  const int C    = in_sizes[6];           // = NHEADS * HID
  const int HID  = C / NHEADS;
  const int E    = in_sizes[1] / 2;
  const int Etot = E + N;
  const int G    = 64;                    // NUM_GRAPHS

  const int* srcA = ei;
  const int* dstA = ei + E;

  // workspace layout (floats)
  float* buf0 = (float*)d_ws;                     // hW (every layer)
  float* buf1 = buf0 + (size_t)N * C;             // rotating features/agg
  float* buf2 = buf1 + (size_t)N * C;
  float* sArr = buf2 + (size_t)N * C;
  float* dArr = sArr + (size_t)N * NHEADS;
  float* den  = dArr + (size_t)N * NHEADS;
  unsigned* mb = (unsigned*)(den + (size_t)N * NHEADS);
  float* eBuf = (float*)(mb + (size_t)N * NHEADS);
  float* sums = eBuf + (size_t)Etot * NHEADS;
  float* cnt  = sums + (size_t)G * C;
  float* tmp  = cnt + G;

  float* outScalar = (float*)d_out;               // [G]
  float* hp        = outScalar + G;               // [G, C]

  const size_t nC = (size_t)N * C;
  const int edgeThreads = Etot * NHEADS;

  const float* inF = x;
  int Kdim = Fin;
  float* hW   = buf0;
  float* aggB = buf1;

  for (int l = 0; l < 3; ++l) {
    dim3 gg((unsigned)((N + 16 * MT - 1) / (16 * MT)), (unsigned)((C + 63) / 64));
    gemm_f32_wmma<<<gg, 128, 0, stream>>>(inF, W[l], hW, N, Kdim, C);

    int waves = N * NHEADS;
    attn_sd<<<(waves * 32 + 255) / 256, 256, 0, stream>>>(hW, aS[l], aD[l],
                                                          sArr, dArr, N, HID);
    init_layer<<<(unsigned)((nC + 255) / 256), 256, 0, stream>>>(
        aggB, mb, den, nC, (size_t)N * NHEADS);

    edge_p1<<<(edgeThreads + 255) / 256, 256, 0, stream>>>(srcA, dstA, sArr,
                                                           dArr, eBuf, mb, E, N);
    edge_p2<<<(edgeThreads + 255) / 256, 256, 0, stream>>>(dstA, eBuf, mb, den,
                                                           E, N);
    edge_p3<<<Etot, 256, 0, stream>>>(srcA, dstA, hW, eBuf, den, aggB, E, N, C,
                                      HID);
    finish_layer<<<(unsigned)((nC + 255) / 256), 256, 0, stream>>>(
        aggB, bias[l], (l > 0 ? inF : nullptr), nC, C);

    // rotate buffers: new features = aggB; next agg reuses the old input buf
    float* nextAgg = (l == 0) ? buf2 : (float*)inF;
    inF = aggB;
    aggB = nextAgg;
    Kdim = C;
  }

  // global mean pool
  zero_f<<<(unsigned)(((size_t)G * C + 255) / 256), 256, 0, stream>>>(sums, (size_t)G * C);
  zero_f<<<1, 64, 0, stream>>>(cnt, (size_t)G);
  pool_cnt<<<(N + 255) / 256, 256, 0, stream>>>(batch, cnt, N);
  pool_add<<<(unsigned)((nC + 255) / 256), 256, 0, stream>>>(inF, batch, sums, N, C);
  pool_div<<<(G * C + 255) / 256, 256, 0, stream>>>(sums, cnt, hp, G, C);

  // MLP head: tmp = hp @ cW1 (WMMA), then relu(+cb1) . cW2 + cb2
  dim3 gm((unsigned)((G + 16 * MT - 1) / (16 * MT)), (unsigned)((HID + 63) / 64));
  gemm_f32_wmma<<<gm, 128, 0, stream>>>(hp, cW1, tmp, G, C, HID);
  mlp_tail<<<G, 32, 0, stream>>>(tmp, cb1, cW2, cb2, outScalar, HID);
}